// fft2D_4277787427262
// MI455X (gfx1250) — compile-verified
//
#include <hip/hip_runtime.h>

// CDNA5 / gfx1250: wave32, f32 WMMA 16x16x4.
typedef float v2f __attribute__((ext_vector_type(2)));
typedef float v8f __attribute__((ext_vector_type(8)));

#define NDIM    64
#define LSTR    76          // LDS row stride in floats (bank-conflict tuned, 16B aligned)
#define NBATCH  8192
#define PI_32   0.09817477042468103f   // pi/32

// D = A(16x4) * B(4x16) + C  accumulate wrapper
static __device__ __forceinline__ v8f wmma_f32(v2f a, v2f b, v8f c) {
  return __builtin_amdgcn_wmma_f32_16x16x4_f32(false, a, false, b, (short)0, c,
                                               false, false);
}

__global__ __launch_bounds__(128) void fft2d64_wmma_kernel(
    const float* __restrict__ x, float* __restrict__ out, int nbatch) {
  // 5 matrices of 64 rows x LSTR stride
  __shared__ float sWR[NDIM * LSTR];
  __shared__ float sWI[NDIM * LSTR];
  __shared__ float sX [NDIM * LSTR];
  __shared__ float sTR[NDIM * LSTR];
  __shared__ float sTI[NDIM * LSTR];

  const int tid  = threadIdx.x;
  const int lane = tid & 31;
  const int wave = tid >> 5;     // 0..3, owns M rows [wave*16, wave*16+16)
  const int half = lane >> 4;    // 0/1
  const int l16  = lane & 15;
  const int m0   = wave * 16;

  // ---- Twiddles: W[j][k] = exp(-2*pi*i*j*k/64), periodic in (j*k) mod 64 ----
  for (int i = tid; i < NDIM * NDIM; i += 128) {
    int j = i >> 6, k = i & 63;
    int p = (j * k) & 63;                 // exact argument reduction
    float s, c;
    __sincosf((float)p * PI_32, &s, &c);
    sWR[j * LSTR + k] = c;
    sWI[j * LSTR + k] = -s;
  }
  __syncthreads();

  for (int b = blockIdx.x; b < nbatch; b += gridDim.x) {
    const float* xb = x + (size_t)b * (NDIM * NDIM);

    // ---- Stage X (16 KB contiguous) into LDS, float4 coalesced ----
    for (int i = tid; i < (NDIM * NDIM) / 4; i += 128) {
      int r  = i >> 4;               // 16 float4 per row
      int c4 = (i & 15) << 2;
      float4 v = ((const float4*)xb)[i];
      *(float4*)&sX[r * LSTR + c4] = v;
    }
    // Prefetch next batch's X into L2 while we compute (global_prefetch_b8)
    {
      int bn = b + gridDim.x;
      if (bn < nbatch)
        __builtin_prefetch(x + (size_t)bn * (NDIM * NDIM) + tid * 32, 0, 1);
    }
    __syncthreads();

    // =========== Stage 1: T = X * W  (real A, complex B) ===========
    // A frag (16x4):  a[v] = X[m0 + l16][4*ks + v + 2*half]      -> contiguous v2f
    // B frag (4x16):  b[v] = W[4*ks + v + 2*half][n]  = W[n][...] (symmetry)
    v8f tR[4], tI[4];
    const v8f vzero = {0.f, 0.f, 0.f, 0.f, 0.f, 0.f, 0.f, 0.f};
#pragma unroll
    for (int t = 0; t < 4; ++t) { tR[t] = vzero; tI[t] = vzero; }

    for (int ks = 0; ks < 16; ++ks) {
      const int kb = 4 * ks + 2 * half;
      v2f a = *(const v2f*)&sX[(m0 + l16) * LSTR + kb];
#pragma unroll
      for (int t = 0; t < 4; ++t) {
        const int n = t * 16 + l16;
        v2f br = *(const v2f*)&sWR[n * LSTR + kb];  // W symmetric
        v2f bi = *(const v2f*)&sWI[n * LSTR + kb];
        tR[t] = wmma_f32(a, br, tR[t]);
        tI[t] = wmma_f32(a, bi, tI[t]);
      }
    }

    // C/D layout: c[v] = D[m0 + v + 8*half][t*16 + l16]
#pragma unroll
    for (int t = 0; t < 4; ++t) {
      const int col = t * 16 + l16;
#pragma unroll
      for (int v = 0; v < 8; ++v) {
        const int row = m0 + v + 8 * half;
        sTR[row * LSTR + col] = tR[t][v];
        sTI[row * LSTR + col] = tI[t][v];
      }
    }
    __syncthreads();

    // =========== Stage 2: r2 = W * T  (complex x complex) ===========
    v8f rR[4], rI[4];
#pragma unroll
    for (int t = 0; t < 4; ++t) { rR[t] = vzero; rI[t] = vzero; }

    for (int ks = 0; ks < 16; ++ks) {
      const int kb = 4 * ks + 2 * half;
      v2f ar  = *(const v2f*)&sWR[(m0 + l16) * LSTR + kb];
      v2f ai  = *(const v2f*)&sWI[(m0 + l16) * LSTR + kb];
      v2f ain = {-ai.x, -ai.y};   // f32 WMMA NEG bits are C-only -> negate in VALU
#pragma unroll
      for (int t = 0; t < 4; ++t) {
        const int n = t * 16 + l16;
        v2f br, bi;
        br.x = sTR[(kb + 0) * LSTR + n];
        br.y = sTR[(kb + 1) * LSTR + n];
        bi.x = sTI[(kb + 0) * LSTR + n];
        bi.y = sTI[(kb + 1) * LSTR + n];
        rR[t] = wmma_f32(ar,  br, rR[t]);
        rR[t] = wmma_f32(ain, bi, rR[t]);
        rI[t] = wmma_f32(ar,  bi, rI[t]);
        rI[t] = wmma_f32(ai,  br, rI[t]);
      }
    }

    // ---- Store interleaved {re, im}: out[b][ky][kx][2] ----
    float* ob = out + (size_t)b * (NDIM * NDIM * 2);
#pragma unroll
    for (int t = 0; t < 4; ++t) {
      const int kx = t * 16 + l16;
#pragma unroll
      for (int v = 0; v < 8; ++v) {
        const int ky = m0 + v + 8 * half;
        float2 val = {rR[t][v], rI[t][v]};
        *(float2*)&ob[(ky * NDIM + kx) * 2] = val;
      }
    }
    __syncthreads();  // protect sX/sTR/sTI before next batch overwrites
  }
}

extern "C" void kernel_launch(void* const* d_in, const int* in_sizes, int n_in,
                              void* d_out, int out_size, void* d_ws, size_t ws_size,
                              hipStream_t stream) {
  (void)in_sizes; (void)n_in; (void)d_ws; (void)ws_size; (void)out_size;
  const float* x = (const float*)d_in[0];
  float* out = (float*)d_out;
  // 2048 blocks x 128 threads: 4 batches per block, 3 blocks/WGP by LDS (95 KB)
  fft2d64_wmma_kernel<<<dim3(2048), dim3(128), 0, stream>>>(x, out, NBATCH);
}